// get_loss_438086664395
// MI455X (gfx1250) — compile-verified
//
#include <hip/hip_runtime.h>
#include <hip/hip_bf16.h>

typedef __attribute__((ext_vector_type(2))) float v2f;
typedef __attribute__((ext_vector_type(8))) float v8f;

#define BATCHES   16
#define NPTS      65536
#define NCLS      5
#define NSTRUCT   4
#define AW        200.0f

#define NB1       32      // blocks per batch for phase 1
#define T1        256
#define NB2       512     // blocks for phase 2
#define T2        256
#define M_ROWS    1048576LL

// ws layout (floats)
#define WS_PART1   0                       // [16*NB1][16]
#define WS_WMAT    (BATCHES * NB1 * 16)    // [16][8]
#define WS_PART2   (WS_WMAT + BATCHES * 8) // [NB2][16] (10 used)

__device__ __forceinline__ float wave_sum32(float v) {
#pragma unroll
    for (int off = 16; off > 0; off >>= 1) v += __shfl_xor(v, off, 32);
    return v;
}

// ---------------- Phase 1: per-batch class stats (deterministic) ----------------
__global__ void k1_batch_stats(const float* __restrict__ pred,
                               const int* __restrict__ target,
                               const float* __restrict__ points,
                               float* __restrict__ part1) {
    const int b   = blockIdx.y;
    const int blk = blockIdx.x;
    const int tid = threadIdx.x;
    const int rowsPerBlock = NPTS / NB1; // 2048
    const long long baseRow = (long long)b * NPTS + (long long)blk * rowsPerBlock;

    // acc: [0..3] gt_cnt, [4..7] pr_cnt, [8..11] gt_zsum, [12..15] pr_zsum
    float acc[16];
#pragma unroll
    for (int i = 0; i < 16; ++i) acc[i] = 0.0f;

    for (int r = tid; r < rowsPerBlock; r += T1) {
        const long long row = baseRow + r;
        const float* p = pred + row * NCLS;
        float x0 = p[0], x1 = p[1], x2 = p[2], x3 = p[3], x4 = p[4];
        // argmax, first index wins ties (strict >)
        int lab = 0; float best = x0;
        if (x1 > best) { best = x1; lab = 1; }
        if (x2 > best) { best = x2; lab = 2; }
        if (x3 > best) { best = x3; lab = 3; }
        if (x4 > best) { best = x4; lab = 4; }
        const int   t = target[row];
        const float z = points[row * 9 + 2];
#pragma unroll
        for (int c = 0; c < NSTRUCT; ++c) {
            const float isGT = (t   == c) ? 1.0f : 0.0f;
            const float isPR = (lab == c) ? 1.0f : 0.0f;
            acc[c]      += isGT;
            acc[4 + c]  += isPR;
            acc[8 + c]  += isGT * z;
            acc[12 + c] += isPR * z;
        }
    }

    __shared__ float red[T1 / 32][16];
    const int wave = tid >> 5, lane = tid & 31;
#pragma unroll
    for (int i = 0; i < 16; ++i) {
        const float s = wave_sum32(acc[i]);
        if (lane == 0) red[wave][i] = s;
    }
    __syncthreads();
    if (tid < 16) {
        float s = 0.0f;
#pragma unroll
        for (int w = 0; w < T1 / 32; ++w) s += red[w][tid];
        part1[(((long long)b * NB1 + blk) << 4) + tid] = s;
    }
}

// ---------------- Phase 1 finalize: weights w[16][5] ----------------
__global__ void k2_weights(const float* __restrict__ part1,
                           float* __restrict__ wMat) {
    __shared__ float sts[BATCHES][16];
    const int t = threadIdx.x;            // 256
    const int b = t >> 4, s = t & 15;
    float sum = 0.0f;
    for (int blk = 0; blk < NB1; ++blk)
        sum += part1[(((long long)b * NB1 + blk) << 4) + s];
    sts[b][s] = sum;
    __syncthreads();

    if (t < BATCHES) {
        const int bb = t;
        float gc[4], pc[4], gm[4], pm[4];
        bool GT[4], PR[4];
#pragma unroll
        for (int c = 0; c < 4; ++c) {
            gc[c] = sts[bb][c];
            pc[c] = sts[bb][4 + c];
            gm[c] = sts[bb][8 + c]  / fmaxf(gc[c], 1.0f);
            pm[c] = sts[bb][12 + c] / fmaxf(pc[c], 1.0f);
            GT[c] = gc[c] > 0.0f;
            PR[c] = pc[c] > 0.0f;
        }
        float w[5] = {1.f, 1.f, 1.f, 1.f, 1.f};
#pragma unroll
        for (int c = 0; c < 4; ++c)
            if (!GT[c] && PR[c]) w[c] += AW;
        const int   pi[6] = {0, 1, 2, 0, 1, 0};
        const int   pj[6] = {1, 2, 3, 2, 3, 3};
        const float pmul[6] = {1.f, 1.f, 1.f, 2.f, 2.f, 3.f};
#pragma unroll
        for (int q = 0; q < 6; ++q) {
            const int i = pi[q], j = pj[q];
            const bool cond = GT[i] && PR[i] && GT[j] && PR[j] &&
                              (gm[i] < gm[j]) && (pm[i] >= pm[j]);
            if (cond) { const float add = pmul[q] * AW; w[i] += add; w[j] += add; }
        }
#pragma unroll
        for (int c = 0; c < 5; ++c) wMat[bb * 8 + c] = w[c];
#pragma unroll
        for (int c = 5; c < 8; ++c) wMat[bb * 8 + c] = 0.0f;
    }
}

// ---------------- Phase 2: NLL segment sums (deterministic) ----------------
__global__ void k3_nll(const float* __restrict__ pred_prev,
                       const int* __restrict__ tgt_prev,
                       float* __restrict__ part2) {
    const int tid = threadIdx.x;
    const long long start  = (long long)blockIdx.x * T2 + tid;
    const long long stride = (long long)NB2 * T2;  // 131072

    float sAcc[5] = {0, 0, 0, 0, 0};
    float nAcc[5] = {0, 0, 0, 0, 0};

    for (long long row = start; row < M_ROWS; row += stride) {
        const float* p = pred_prev + row * NCLS;
        __builtin_prefetch(pred_prev + (row + stride) * NCLS, 0, 0);
        float x0 = p[0], x1 = p[1], x2 = p[2], x3 = p[3], x4 = p[4];
        const float m = fmaxf(fmaxf(fmaxf(x0, x1), fmaxf(x2, x3)), x4);
        const float sum = __expf(x0 - m) + __expf(x1 - m) + __expf(x2 - m) +
                          __expf(x3 - m) + __expf(x4 - m);
        const float lse = m + __logf(sum);
        const int t = tgt_prev[row];
        const float xt = (t == 0) ? x0 : (t == 1) ? x1 : (t == 2) ? x2
                       : (t == 3) ? x3 : x4;
        const float nll = lse - xt;
#pragma unroll
        for (int c = 0; c < 5; ++c) {
            const float hit = (t == c) ? 1.0f : 0.0f;
            sAcc[c] += hit * nll;
            nAcc[c] += hit;
        }
    }

    float acc[10];
#pragma unroll
    for (int c = 0; c < 5; ++c) { acc[c] = sAcc[c]; acc[5 + c] = nAcc[c]; }

    __shared__ float red[T2 / 32][10];
    const int wave = tid >> 5, lane = tid & 31;
#pragma unroll
    for (int i = 0; i < 10; ++i) {
        const float s = wave_sum32(acc[i]);
        if (lane == 0) red[wave][i] = s;
    }
    __syncthreads();
    if (tid < 10) {
        float s = 0.0f;
#pragma unroll
        for (int w = 0; w < T2 / 32; ++w) s += red[w][tid];
        part2[blockIdx.x * 16 + tid] = s;
    }
}

// ---------------- Finalize: s/n reduce, then WMMA epilogue ----------------
// loss = sum_b (w[b]·s) / (w[b]·n) via V_WMMA_F32_16X16X4_F32:
//   D(16x16) = A(16x4) x B(4x16) + C, two chained WMMAs cover K=5 (padded to 8).
//   B column N=0 carries s, column N=1 carries n.
__global__ void k4_final(const float* __restrict__ part2,
                         const float* __restrict__ wMat,
                         float* __restrict__ out) {
    __shared__ float sn[16];        // [0..4]=s, [5..9]=n
    __shared__ float dls[32 * 8];
    const int lane = threadIdx.x;   // exactly 32 threads (one wave)

    if (lane < 16) {
        float sum = 0.0f;
        if (lane < 10)
            for (int p = 0; p < NB2; ++p) sum += part2[p * 16 + lane];
        sn[lane] = sum;
    }
    __syncthreads();

    const int  m  = lane & 15;
    const bool hi = lane >= 16;

    // A 16x4 f32: lanes 0-15 -> K0,K1 ; lanes 16-31 -> K2,K3 (M = lane%16)
    v2f a0, a1;
    a0.x = wMat[m * 8 + (hi ? 2 : 0)];
    a0.y = wMat[m * 8 + (hi ? 3 : 1)];
    a1.x = hi ? 0.0f : wMat[m * 8 + 4];   // K=4 in chunk 1, K=5..7 zero-pad
    a1.y = 0.0f;

    // B 4x16 f32: V0 = row K0 (lanes 0-15) / row K2 (lanes 16-31); V1 = K1 / K3
    const int k0 = hi ? 2 : 0, k1 = hi ? 3 : 1;
    v2f b0, b1;
    b0.x = (m == 0) ? sn[k0] : (m == 1) ? sn[5 + k0] : 0.0f;
    b0.y = (m == 0) ? sn[k1] : (m == 1) ? sn[5 + k1] : 0.0f;
    b1.x = (!hi && m == 0) ? sn[4] : (!hi && m == 1) ? sn[9] : 0.0f;
    b1.y = 0.0f;

    v8f c = {};
    c = __builtin_amdgcn_wmma_f32_16x16x4_f32(false, a0, false, b0,
                                              (short)0, c, false, false);
    c = __builtin_amdgcn_wmma_f32_16x16x4_f32(false, a1, false, b1,
                                              (short)0, c, false, false);

#pragma unroll
    for (int v = 0; v < 8; ++v) dls[lane * 8 + v] = c[v];
    __syncthreads();

    if (lane == 0) {
        // D col 0 (= w@s): lane 0 has M=0..7, lane 16 has M=8..15
        // D col 1 (= w@n): lane 1 has M=0..7, lane 17 has M=8..15
        float loss = 0.0f;
#pragma unroll
        for (int mm = 0; mm < 8; ++mm) loss += dls[0 * 8 + mm]  / dls[1 * 8 + mm];
#pragma unroll
        for (int mm = 0; mm < 8; ++mm) loss += dls[16 * 8 + mm] / dls[17 * 8 + mm];
        out[0] = loss;
    }
}

extern "C" void kernel_launch(void* const* d_in, const int* in_sizes, int n_in,
                              void* d_out, int out_size, void* d_ws, size_t ws_size,
                              hipStream_t stream) {
    const float* pred      = (const float*)d_in[0]; // [16,65536,5]
    const int*   target    = (const int*)d_in[1];   // [16,65536]
    const float* points    = (const float*)d_in[2]; // [16,65536,9]
    const float* pred_prev = (const float*)d_in[3]; // [1048576,5]
    const int*   tgt_prev  = (const int*)d_in[4];   // [1048576] (int32: jax x64 off)
    float* out = (float*)d_out;

    float* ws    = (float*)d_ws;
    float* part1 = ws + WS_PART1;
    float* wMat  = ws + WS_WMAT;
    float* part2 = ws + WS_PART2;

    k1_batch_stats<<<dim3(NB1, BATCHES), T1, 0, stream>>>(pred, target, points, part1);
    k2_weights<<<1, 256, 0, stream>>>(part1, wMat);
    k3_nll<<<NB2, T2, 0, stream>>>(pred_prev, tgt_prev, part2);
    k4_final<<<1, 32, 0, stream>>>(part2, wMat, out);
}